// TimeSeries_12232066859130
// MI455X (gfx1250) — compile-verified
//
#include <hip/hip_runtime.h>
#include <hip/hip_bf16.h>

// ---------------------------------------------------------------------------
// Problem dims (compile-time constants from the reference)
// ---------------------------------------------------------------------------
constexpr int CB  = 16;    // batch
constexpr int CT  = 512;   // time steps
constexpr int CIN = 128;   // input features
constexpr int CH  = 256;   // hidden
constexpr int CD  = 8;     // latent dim
constexpr int CBT = CB * CT; // 8192 flattened (b,t) rows

typedef __attribute__((ext_vector_type(2))) float v2f;
typedef __attribute__((ext_vector_type(4))) float v4f;
typedef __attribute__((ext_vector_type(8))) float v8f;

// D = A(16x4) * B(4x16) + C, full fp32 WMMA (CDNA5 V_WMMA_F32_16X16X4_F32)
__device__ __forceinline__ v8f wmma_f32(v2f a, v2f b, v8f c) {
    return __builtin_amdgcn_wmma_f32_16x16x4_f32(
        /*neg_a=*/false, a, /*neg_b=*/false, b,
        /*c_mod=*/(short)0, c, /*reuse_a=*/false, /*reuse_b=*/false);
}

// ---------------------------------------------------------------------------
// Kernel 1: x_proj = y @ W_ih^T + b_ih      (8192x128) x (128x256) -> (8192x256)
// One wave per 16x16 output tile; K = 128 = 32 WMMA_F32_16X16X4 steps.
// ---------------------------------------------------------------------------
__global__ void xproj_wmma_kernel(const float* __restrict__ y,
                                  const float* __restrict__ W_ih,
                                  const float* __restrict__ b_ih,
                                  float* __restrict__ xproj) {
    const int gwave = (blockIdx.x * blockDim.x + threadIdx.x) >> 5;
    const int lane  = threadIdx.x & 31;
    const int NT    = CH / 16;                 // 16 N-tiles
    const int mt    = gwave / NT;
    const int nt    = gwave % NT;
    if (mt >= CBT / 16) return;
    const int m0  = mt * 16;
    const int n0  = nt * 16;
    const int row = lane & 15;
    const int hi  = lane >> 4;                 // 0/1 -> K offset 0/2

    v8f acc = {};
    #pragma unroll
    for (int k = 0; k < CIN; k += 4) {
        v2f a, b;
        const int kk = k + 2 * hi;
        a.x = y[(m0 + row) * CIN + kk + 0];
        a.y = y[(m0 + row) * CIN + kk + 1];
        b.x = W_ih[(n0 + row) * CIN + kk + 0];   // B(k,n) = W_ih[n][k]
        b.y = W_ih[(n0 + row) * CIN + kk + 1];
        acc = wmma_f32(a, b, acc);
    }
    const float bias = b_ih[n0 + row];
    #pragma unroll
    for (int r = 0; r < 8; ++r) {
        const int m = m0 + r + 8 * hi;
        xproj[m * CH + n0 + row] = acc[r] + bias;
    }
}

// ---------------------------------------------------------------------------
// Kernel 2: sequential recurrence, one persistent workgroup (16 waves).
//   h_new = tanh(x_t + h @ W_hh^T + b_hh)
// Wave w owns hidden columns [16w, 16w+16): its 64 B-fragments of W_hh stay in
// registers (128 VGPRs); h is double-buffered in LDS (2 x 16KB); one barrier
// per time step.
// ---------------------------------------------------------------------------
__global__ void __launch_bounds__(512)
rnn_wmma_kernel(const float* __restrict__ xproj,   // (B,T,H)
                const float* __restrict__ W_hh,    // (H,H) [n][k]
                const float* __restrict__ b_hh,    // (H)
                float* __restrict__ enc) {         // (B,T,H)
    __shared__ float hbuf[2][CB * CH];             // 2 x 16 KB

    const int wv   = threadIdx.x >> 5;             // 0..15 -> N-tile
    const int lane = threadIdx.x & 31;
    const int row  = lane & 15;
    const int hi   = lane >> 4;
    const int n    = wv * 16 + row;                // hidden column owned

    // Pin W_hh B-fragments in registers: B(k,n) = W_hh[n][k], K = 256 -> 64 frags
    v2f wb[64];
    #pragma unroll
    for (int kc = 0; kc < 64; ++kc) {
        const int k = 4 * kc + 2 * hi;
        wb[kc].x = W_hh[n * CH + k + 0];
        wb[kc].y = W_hh[n * CH + k + 1];
    }
    const float bias = b_hh[n];

    for (int i = threadIdx.x; i < CB * CH; i += 512) hbuf[0][i] = 0.0f;
    __syncthreads();

    int cur = 0;
    for (int t = 0; t < CT; ++t) {
        // C init = x_t + b_hh, laid out per the 16x16 f32 C fragment
        v8f acc;
        #pragma unroll
        for (int r = 0; r < 8; ++r) {
            const int bb = r + 8 * hi;                    // batch row (M)
            acc[r] = xproj[(bb * CT + t) * CH + n] + bias;
        }
        // acc += h(16x256) @ W_hh^T tile: 64 K-chunks of 4
        const float* hrow = &hbuf[cur][0];
        #pragma unroll
        for (int kc = 0; kc < 64; ++kc) {
            v2f a;
            const int k = 4 * kc + 2 * hi;
            a.x = hrow[row * CH + k + 0];                 // A(m,k) = h[m][k], m=row
            a.y = hrow[row * CH + k + 1];
            acc = wmma_f32(a, wb[kc], acc);
        }
        const int nxt = cur ^ 1;
        #pragma unroll
        for (int r = 0; r < 8; ++r) {
            const int bb = r + 8 * hi;
            const float hv = tanhf(acc[r]);
            hbuf[nxt][bb * CH + n] = hv;
            enc[(bb * CT + t) * CH + n] = hv;
        }
        __syncthreads();
        cur = nxt;
    }
}

// ---------------------------------------------------------------------------
// Kernel 3: heads.  Wcat rows = [W_mean(8); W_bd(16)] -> 24 output cols.
//   cols 0..7  -> mean (written straight to d_out)
//   cols 8..23 -> bd   (workspace, consumed by prec kernel)
// One wave per 16x16 tile; 2 N-tiles (second tile half-masked), K = 256.
// ---------------------------------------------------------------------------
__global__ void head_wmma_kernel(const float* __restrict__ enc,
                                 const float* __restrict__ W_mean,
                                 const float* __restrict__ b_mean,
                                 const float* __restrict__ W_bd,
                                 const float* __restrict__ b_bd,
                                 float* __restrict__ mean_out,   // (B*T, 8)
                                 float* __restrict__ bd_ws) {    // (B*T, 16)
    const int gwave = (blockIdx.x * blockDim.x + threadIdx.x) >> 5;
    const int lane  = threadIdx.x & 31;
    const int mt    = gwave >> 1;
    const int nt    = gwave & 1;
    if (mt >= CBT / 16) return;
    const int m0  = mt * 16;
    const int n0  = nt * 16;
    const int row = lane & 15;
    const int hi  = lane >> 4;
    const int n   = n0 + row;                       // 0..31; valid < 24

    v8f acc = {};
    #pragma unroll 8
    for (int k = 0; k < CH; k += 4) {
        v2f a, b;
        const int kk = k + 2 * hi;
        a.x = enc[(m0 + row) * CH + kk + 0];
        a.y = enc[(m0 + row) * CH + kk + 1];
        if (n < CD) {                               // W_mean rows
            b.x = W_mean[n * CH + kk + 0];
            b.y = W_mean[n * CH + kk + 1];
        } else if (n < CD + 2 * CD) {               // W_bd rows
            b.x = W_bd[(n - CD) * CH + kk + 0];
            b.y = W_bd[(n - CD) * CH + kk + 1];
        } else {
            b.x = 0.0f; b.y = 0.0f;
        }
        acc = wmma_f32(a, b, acc);
    }
    const float bias = (n < CD) ? b_mean[n] : ((n < 3 * CD) ? b_bd[n - CD] : 0.0f);
    #pragma unroll
    for (int r = 0; r < 8; ++r) {
        const int m = m0 + r + 8 * hi;
        const float v = acc[r] + bias;
        if (n < CD)            mean_out[m * CD + n] = v;
        else if (n < 3 * CD)   bd_ws[m * 16 + (n - CD)] = v;
    }
}

// ---------------------------------------------------------------------------
// Kernel 4: assemble prec (16,8,512,512) tridiagonal tensor. Pure bandwidth:
// one v4f (4 consecutive j) per thread, nontemporal stores (134 MB,
// ~5.7 us floor at 23.3 TB/s). bd_ws cols 0..7 = diag_e, 8..15 = off_e.
// ---------------------------------------------------------------------------
__global__ void prec_kernel(const float* __restrict__ bd_ws,
                            float* __restrict__ prec) {
    const long long q = (long long)blockIdx.x * blockDim.x + threadIdx.x;
    const int QR = CT / 4;                           // 128 quads per row
    const int jq   = (int)(q % QR);
    const long long rowi = q / QR;
    const int i  = (int)(rowi % CT);
    const int rd = (int)(rowi / CT);                 // b*8 + d
    const int d  = rd % CD;
    const int b  = rd / CD;
    const int j0 = jq * 4;

    v4f out = {0.f, 0.f, 0.f, 0.f};
    if (i >= j0 - 1 && i <= j0 + 4) {                // band intersects this quad
        #pragma unroll
        for (int c = 0; c < 4; ++c) {
            const int j = j0 + c;
            float val = 0.0f;
            if (j == i) {
                const float de = bd_ws[(b * CT + i) * 16 + d];
                const float oe = (i > 0) ? bd_ws[(b * CT + i - 1) * 16 + 8 + d] : 0.0f;
                val = de * de + oe * oe + 1e-5f;
            } else if (j == i + 1) {                 // super-diag: supd[i]
                val = bd_ws[(b * CT + i) * 16 + d] * bd_ws[(b * CT + i) * 16 + 8 + d];
            } else if (i == j + 1) {                 // sub-diag: supd[j]
                val = bd_ws[(b * CT + j) * 16 + d] * bd_ws[(b * CT + j) * 16 + 8 + d];
            }
            out[c] = val;
        }
    }
    __builtin_nontemporal_store(out, ((v4f*)prec) + q);
}

// ---------------------------------------------------------------------------
// Launch
// ---------------------------------------------------------------------------
extern "C" void kernel_launch(void* const* d_in, const int* in_sizes, int n_in,
                              void* d_out, int out_size, void* d_ws, size_t ws_size,
                              hipStream_t stream) {
    const float* y      = (const float*)d_in[0];  // (16,512,128)
    const float* W_ih   = (const float*)d_in[1];  // (256,128)
    const float* W_hh   = (const float*)d_in[2];  // (256,256)
    const float* b_ih   = (const float*)d_in[3];  // (256)
    const float* b_hh   = (const float*)d_in[4];  // (256)
    const float* W_mean = (const float*)d_in[5];  // (8,256)
    const float* b_mean = (const float*)d_in[6];  // (8)
    const float* W_bd   = (const float*)d_in[7];  // (16,256)
    const float* b_bd   = (const float*)d_in[8];  // (16)

    float* mean_out = (float*)d_out;              // 16*512*8 = 65536 floats
    float* prec_out = mean_out + CB * CT * CD;    // 16*8*512*512 floats

    float* xproj = (float*)d_ws;                  // 8192*256
    float* enc   = xproj + CBT * CH;              // 8192*256
    float* bd_ws = enc + CBT * CH;                // 8192*16

    // K1: x_proj. 8192 tiles, 1 wave each -> 262144 threads
    xproj_wmma_kernel<<<(CBT / 16) * (CH / 16) * 32 / 256, 256, 0, stream>>>(
        y, W_ih, b_ih, xproj);

    // K2: persistent recurrence, single workgroup of 16 waves
    rnn_wmma_kernel<<<1, 512, 0, stream>>>(xproj, W_hh, b_hh, enc);

    // K3: heads. 512 M-tiles x 2 N-tiles = 1024 waves -> 32768 threads
    head_wmma_kernel<<<(CBT / 16) * 2 * 32 / 256, 256, 0, stream>>>(
        enc, W_mean, b_mean, W_bd, b_bd, mean_out, bd_ws);

    // K4: tridiagonal assembly. 33.55M elems / 4 per thread = 8,388,608 threads
    const long long quads = (long long)CB * CD * CT * (CT / 4);
    prec_kernel<<<(int)(quads / 256), 256, 0, stream>>>(bd_ws, prec_out);
}